// Prototype_47614007444058
// MI455X (gfx1250) — compile-verified
//
#include <hip/hip_runtime.h>

typedef float v2f __attribute__((ext_vector_type(2)));
typedef float v8f __attribute__((ext_vector_type(8)));

#define N_CLASS  100
#define N_CENTER 4
#define HIDDEN   128
#define BATCH    2048
#define CHUNK    32

// ---------------- Phase 1: deterministic per-class bucketing ----------------

__global__ void zero_counts_kernel(int* counts) {
    int t = threadIdx.x;
    if (t < N_CLASS) counts[t] = 0;
}

__global__ void hist_kernel(const int* __restrict__ y, int* counts) {
    int b = blockIdx.x * blockDim.x + threadIdx.x;
    if (b < BATCH) atomicAdd(&counts[y[b]], 1);
}

__global__ void scan_kernel(const int* __restrict__ counts, int* offsets) {
    // single thread: 100-element exclusive scan (negligible work)
    int acc = 0;
    for (int k = 0; k < N_CLASS; ++k) {
        offsets[k] = acc;
        acc += counts[k];
    }
    offsets[N_CLASS] = acc;
}

// Deterministic stable scatter: position = class offset + rank among earlier
// samples of the same class. No atomics -> identical order on every call.
__global__ void scatter_kernel(const int* __restrict__ y,
                               const int* __restrict__ offsets,
                               int* __restrict__ idxbuf) {
    int b = blockIdx.x * blockDim.x + threadIdx.x;
    if (b >= BATCH) return;
    int k = y[b];
    int rank = 0;
    for (int i = 0; i < b; ++i) rank += (y[i] == k) ? 1 : 0;
    idxbuf[offsets[k] + rank] = b;
}

// ---------------- Phase 2: 400 Gram matrices via f32 WMMA ----------------
// One workgroup per (class, center). 8 waves; wave w owns output tile-row w
// (rows w*16..w*16+15 of the 128x128 Gram matrix), 8 tiles of 16x16 each.
//
// LDS holds the diff chunk with row-pairs interleaved:
//   element (k, col) -> sd[((k>>1)*HIDDEN + col)*2 + (k&1)]
// so every WMMA A/B operand (rows kb'+0 / kb'+1, same col, kb' even) is one
// aligned 8-byte ds_load_b64 into a consecutive VGPR pair -> no packing moves.

__global__ __launch_bounds__(256)
void gram_kernel(const float* __restrict__ x,
                 const float* __restrict__ mu,      // class_mu[0] : [4][128]
                 const float* __restrict__ cov_in,  // [100][4][128][128]
                 const int*   __restrict__ offsets, // [101]
                 const int*   __restrict__ idxbuf,  // [2048]
                 float*       __restrict__ out)     // [100][4][128][128]
{
    __shared__ float sd[CHUNK * HIDDEN];            // 16 KB diff staging

    const int k    = blockIdx.x >> 2;   // class
    const int c    = blockIdx.x & 3;    // center
    const int t    = threadIdx.x;
    const int lane = t & 31;
    const int wave = t >> 5;            // 0..7 : i-tile row block
    const int hi   = lane >> 4;         // lane group (0: lanes 0-15, 1: 16-31)
    const int lo   = lane & 15;

    const int start = offsets[k];
    const int n_k   = offsets[k + 1] - start;

    v8f acc[8];
    #pragma unroll
    for (int j = 0; j < 8; ++j)
        #pragma unroll
        for (int e = 0; e < 8; ++e) acc[j][e] = 0.0f;

    const v2f* __restrict__ sdp = (const v2f*)sd;   // pair-indexed view

    for (int cs = 0; cs < n_k; cs += CHUNK) {
        // Cooperative load of up to CHUNK diff rows into LDS, zero-padded.
        // Zero rows contribute zero outer products -> no tail masking needed.
        #pragma unroll
        for (int e = 0; e < (CHUNK * HIDDEN) / 256; ++e) {
            int linear = e * 256 + t;
            int row = linear >> 7;      // / HIDDEN
            int col = linear & (HIDDEN - 1);
            float v = 0.0f;
            if (cs + row < n_k) {
                int b = idxbuf[start + cs + row];
                v = x[b * HIDDEN + col] - mu[c * HIDDEN + col];
            }
            sd[((row >> 1) * HIDDEN + col) * 2 + (row & 1)] = v;
        }
        __syncthreads();

        // K-steps of 4 batch rows each; A (16x4, col block = wave) and
        // B (4x16, col block = j) share the same lane/VGPR index function:
        //   operand = { D[kb + 2*hi][col0 + lo], D[kb + 2*hi + 1][col0 + lo] }
        // = one b64 load at pair index (kb/2 + hi)*HIDDEN + col0 + lo.
        #pragma unroll
        for (int kk = 0; kk < CHUNK / 4; ++kk) {
            const int prow = (kk * 2 + hi) * HIDDEN;   // (kb>>1 + hi) * HIDDEN
            v2f a = sdp[prow + wave * 16 + lo];
            #pragma unroll
            for (int j = 0; j < 8; ++j) {
                v2f bm = sdp[prow + j * 16 + lo];
                acc[j] = __builtin_amdgcn_wmma_f32_16x16x4_f32(
                    false, a, false, bm, (short)0, acc[j], false, false);
            }
        }
        __syncthreads();
    }

    // Store: D-matrix VGPR r holds row r (lanes 0-15) / row r+8 (lanes 16-31).
    const size_t base = (size_t)(k * N_CENTER + c) * HIDDEN * HIDDEN;
    float*       ob = out + base;
    const float* cb = cov_in + base;
    #pragma unroll
    for (int j = 0; j < 8; ++j) {
        #pragma unroll
        for (int r = 0; r < 8; ++r) {
            int m = wave * 16 + r + 8 * hi;
            int n = j * 16 + lo;
            size_t o = (size_t)m * HIDDEN + n;
            ob[o] = cb[o] + acc[j][r];
        }
    }
}

// ---------------- Launch ----------------

extern "C" void kernel_launch(void* const* d_in, const int* in_sizes, int n_in,
                              void* d_out, int out_size, void* d_ws, size_t ws_size,
                              hipStream_t stream) {
    (void)in_sizes; (void)n_in; (void)out_size; (void)ws_size;

    const float* x   = (const float*)d_in[0];   // [2048][128]
    const int*   y   = (const int*)d_in[1];     // [2048]
    const float* mu  = (const float*)d_in[2];   // [100][4][128]; only [0] used
    const float* cov = (const float*)d_in[3];   // [100][4][128][128]
    float*       out = (float*)d_out;

    int* ws      = (int*)d_ws;
    int* counts  = ws;          // 100 ints
    int* offsets = ws + 128;    // 101 ints
    int* idxbuf  = ws + 256;    // 2048 ints

    zero_counts_kernel<<<1, 128, 0, stream>>>(counts);
    hist_kernel<<<BATCH / 256, 256, 0, stream>>>(y, counts);
    scan_kernel<<<1, 1, 0, stream>>>(counts, offsets);
    scatter_kernel<<<BATCH / 256, 256, 0, stream>>>(y, offsets, idxbuf);
    gram_kernel<<<N_CLASS * N_CENTER, 256, 0, stream>>>(x, mu, cov, offsets,
                                                        idxbuf, out);
}